// GCN_86466281603301
// MI455X (gfx1250) — compile-verified
//
#include <hip/hip_runtime.h>
#include <hip/hip_bf16.h>

typedef __attribute__((ext_vector_type(2))) float v2f;
typedef __attribute__((ext_vector_type(8))) float v8f;

// ---------------------------------------------------------------------------
// Generic zero-fill
// ---------------------------------------------------------------------------
__global__ void zero_f(float* __restrict__ p, long long n) {
    long long t = (long long)blockIdx.x * blockDim.x + threadIdx.x;
    if (t < n) p[t] = 0.0f;
}

// ---------------------------------------------------------------------------
// Degree count: deg[dst[e]] += 1   (deg pre-zeroed)
// ---------------------------------------------------------------------------
__global__ void count_deg(const int* __restrict__ dst, float* __restrict__ deg, int E) {
    int e = blockIdx.x * blockDim.x + threadIdx.x;
    if (e < E) atomicAdd(&deg[dst[e]], 1.0f);
}

// In place: deg[i] <- 1/sqrt(deg[i] + 1)   (self-loop included)
__global__ void make_dis(float* __restrict__ deg, int n) {
    int i = blockIdx.x * blockDim.x + threadIdx.x;
    if (i < n) deg[i] = rsqrtf(deg[i] + 1.0f);
}

// ---------------------------------------------------------------------------
// Pack per-edge record {src, dst, norm, pad} so the hot scatter kernels do a
// single b128 load per lane instead of 4 scalar loads (src, dst, dis[s], dis[d]).
// Coalesced over edges; the dis[] gathers are L2-resident (0.4 MB).
// ---------------------------------------------------------------------------
__global__ void prep_edges(const int* __restrict__ src, const int* __restrict__ dst,
                           const float* __restrict__ dis, int4* __restrict__ rec, int E) {
    int e = blockIdx.x * blockDim.x + threadIdx.x;
    if (e >= E) return;
    int s = src[e];
    int d = dst[e];
    float norm = dis[s] * dis[d];
    int4 r;
    r.x = s;
    r.y = d;
    r.z = __float_as_int(norm);
    r.w = 0;
    rec[e] = r;
}

// ---------------------------------------------------------------------------
// F32 WMMA GEMM: C[M,N] = A[M,K] * B[K,N], row-major, M%16==0, N%16==0, K%4==0.
// One wave per 16x16 output tile, V_WMMA_F32_16X16X4_F32 over the K loop.
//
// Fragment layouts (ISA 7.12.2, 32-bit 16x4 A / 4x16 B / 16x16 C):
//   A: lane<16 -> {A[m][k0], A[m][k0+1]},  lane>=16 -> {A[m][k0+2], A[m][k0+3]}
//   B: lane<16 -> {B[k0][n], B[k0+1][n]},  lane>=16 -> {B[k0+2][n], B[k0+3][n]}
//   C: VGPR r  -> row (r + 8*(lane>>4)), col (lane&15)
// ---------------------------------------------------------------------------
__global__ void gemm_wmma_f32(const float* __restrict__ A, const float* __restrict__ B,
                              float* __restrict__ C, int M, int K, int N) {
    int wave  = (blockIdx.x * blockDim.x + threadIdx.x) >> 5;
    int lane  = threadIdx.x & 31;
    int tilesN = N >> 4;
    int mtile = wave / tilesN;
    int ntile = wave - mtile * tilesN;
    int mbase = mtile << 4;
    if (mbase >= M) return;            // wave-uniform guard (EXEC stays all-1 for WMMA)
    int nbase = ntile << 4;

    int mn    = lane & 15;             // row of A frag / col of B,C frags
    int klsel = (lane >> 4) << 1;      // 0 or 2: which K pair this half-wave holds

    const float* __restrict__ Arow = A + (size_t)(mbase + mn) * K;

    v8f c = {};
    for (int k0 = 0; k0 < K; k0 += 4) {
        int kb = k0 + klsel;
        v2f a, b;
        a.x = Arow[kb];
        a.y = Arow[kb + 1];
        b.x = B[(size_t)kb * N + nbase + mn];
        b.y = B[(size_t)(kb + 1) * N + nbase + mn];
        c = __builtin_amdgcn_wmma_f32_16x16x4_f32(
                /*neg_a=*/false, a, /*neg_b=*/false, b,
                /*c_mod=*/(short)0, c, /*reuse_a=*/false, /*reuse_b=*/false);
    }

    int rowoff = (lane >> 4) << 3;     // 0 or 8
#pragma unroll
    for (int r = 0; r < 8; ++r)
        C[(size_t)(mbase + rowoff + r) * N + nbase + mn] = c[r];
}

// ---------------------------------------------------------------------------
// Hot scatter, packed-record path. F=32: 1 wave = 1 edge (lane = feature);
// F=64: 2 waves per edge. Per edge-wave: one b128 record load (same cacheline
// across lanes), one coalesced 128B gather of h[s][*], one coalesced 128B
// burst of f32 atomics into agg[d][*].
// ---------------------------------------------------------------------------
template <int F>
__global__ void edge_scatter_packed(const int4* __restrict__ rec, const float* __restrict__ h,
                                    float* __restrict__ agg, int E) {
    long long t = (long long)blockIdx.x * blockDim.x + threadIdx.x;
    int e = (int)(t >> (F == 32 ? 5 : 6));
    if (e >= E) return;
    int f = (int)(t & (F - 1));
    int4 r = rec[e];
    float norm = __int_as_float(r.z);
    atomicAdd(&agg[(size_t)r.y * F + f], h[(size_t)r.x * F + f] * norm);
}

// Fallback scatter (used only if ws_size can't hold the 16B/edge records).
template <int F>
__global__ void edge_scatter(const int* __restrict__ src, const int* __restrict__ dst,
                             const float* __restrict__ dis, const float* __restrict__ h,
                             float* __restrict__ agg, int E) {
    long long t = (long long)blockIdx.x * blockDim.x + threadIdx.x;
    int e = (int)(t >> (F == 32 ? 5 : 6));
    if (e >= E) return;
    int f = (int)(t & (F - 1));
    int s = src[e];
    int d = dst[e];
    float norm = dis[s] * dis[d];
    atomicAdd(&agg[(size_t)d * F + f], h[(size_t)s * F + f] * norm);
}

// ---------------------------------------------------------------------------
// Finalize: agg[i] = act( agg[i] + h[i] * dis[node]^2 + b[f] )
// (self-loop term norm_ii = deg^{-1} = dis^2, then bias, then optional ReLU)
// ---------------------------------------------------------------------------
template <bool RELU, int F>
__global__ void finalize(const float* __restrict__ h, float* __restrict__ agg,
                         const float* __restrict__ dis, const float* __restrict__ b,
                         int n_nodes) {
    long long t = (long long)blockIdx.x * blockDim.x + threadIdx.x;
    long long total = (long long)n_nodes * F;
    if (t >= total) return;
    int node = (int)(t >> (F == 32 ? 5 : 6));
    int f    = (int)(t & (F - 1));
    float di = dis[node];
    float v  = agg[t] + h[t] * di * di + b[f];
    if (RELU) v = v > 0.0f ? v : 0.0f;
    agg[t] = v;
}

// ---------------------------------------------------------------------------
// Launch
// ---------------------------------------------------------------------------
extern "C" void kernel_launch(void* const* d_in, const int* in_sizes, int n_in,
                              void* d_out, int out_size, void* d_ws, size_t ws_size,
                              hipStream_t stream) {
    const float* x   = (const float*)d_in[0];
    const int*   ei  = (const int*)d_in[1];
    const float* W1  = (const float*)d_in[2];
    const float* b1  = (const float*)d_in[3];
    const float* W2  = (const float*)d_in[4];
    const float* b2  = (const float*)d_in[5];
    float* out = (float*)d_out;

    const int FIN  = 64;
    const int FHID = 32;
    const int N = in_sizes[0] / FIN;       // 100000
    const int E = in_sizes[1] / 2;         // 1000000
    const int* src = ei;
    const int* dst = ei + E;

    // Workspace layout (floats): deg/dis[N] | h1[N*32] | agg1[N*32] | h2[N*64] | rec[E*4]
    float* deg  = (float*)d_ws;            // becomes dis after make_dis
    float* h1   = deg  + N;
    float* agg1 = h1   + (size_t)N * FHID;
    float* h2   = agg1 + (size_t)N * FHID;
    int4*  rec  = (int4*)(h2 + (size_t)N * FIN);

    size_t base_bytes   = (size_t)N * (1 + FHID + FHID + FIN) * sizeof(float);
    size_t packed_bytes = base_bytes + (size_t)E * sizeof(int4);
    const bool use_packed = (ws_size >= packed_bytes);

    const int B = 256;
    auto blocks = [](long long n, int b) { return (unsigned)((n + b - 1) / b); };

    // ---- degrees -> dis = (deg+1)^{-1/2} ----
    zero_f<<<blocks(N, B), B, 0, stream>>>(deg, N);
    count_deg<<<blocks(E, B), B, 0, stream>>>(dst, deg, E);
    make_dis<<<blocks(N, B), B, 0, stream>>>(deg, N);
    if (use_packed)
        prep_edges<<<blocks(E, B), B, 0, stream>>>(src, dst, deg, rec, E);

    // ---- layer 1: h1 = x @ W1 (WMMA f32), scatter, self-loop + bias + ReLU ----
    {
        long long waves = (long long)(N / 16) * (FHID / 16);
        gemm_wmma_f32<<<blocks(waves * 32, B), B, 0, stream>>>(x, W1, h1, N, FIN, FHID);
    }
    zero_f<<<blocks((long long)N * FHID, B), B, 0, stream>>>(agg1, (long long)N * FHID);
    if (use_packed)
        edge_scatter_packed<FHID><<<blocks((long long)E * FHID, B), B, 0, stream>>>(rec, h1, agg1, E);
    else
        edge_scatter<FHID><<<blocks((long long)E * FHID, B), B, 0, stream>>>(src, dst, deg, h1, agg1, E);
    finalize<true, FHID><<<blocks((long long)N * FHID, B), B, 0, stream>>>(h1, agg1, deg, b1, N);

    // ---- layer 2: h2 = relu_out @ W2 (WMMA f32), scatter into d_out, + bias ----
    {
        long long waves = (long long)(N / 16) * (FIN / 16);
        gemm_wmma_f32<<<blocks(waves * 32, B), B, 0, stream>>>(agg1, W2, h2, N, FHID, FIN);
    }
    zero_f<<<blocks((long long)N * FIN, B), B, 0, stream>>>(out, (long long)N * FIN);
    if (use_packed)
        edge_scatter_packed<FIN><<<blocks((long long)E * FIN, B), B, 0, stream>>>(rec, h2, out, E);
    else
        edge_scatter<FIN><<<blocks((long long)E * FIN, B), B, 0, stream>>>(src, dst, deg, h2, out, E);
    finalize<false, FIN><<<blocks((long long)N * FIN, B), B, 0, stream>>>(h2, out, deg, b2, N);
}